// Skill_Evolve_Hetero_74689481277961
// MI455X (gfx1250) — compile-verified
//
#include <hip/hip_runtime.h>
#include <hip/hip_bf16.h>

typedef __attribute__((ext_vector_type(2))) float v2f;
typedef __attribute__((ext_vector_type(8))) float v8f;

#define DIM 64

// ---------------------------------------------------------------------------
// 1) degree init to 1.0 (self-loop weight)
__global__ void deg_init_kernel(float* __restrict__ deg, int n) {
    int i = blockIdx.x * blockDim.x + threadIdx.x;
    if (i < n) deg[i] = 1.0f;
}

// 2) deg[col[e]] += w[e]  (L2-resident float atomics)
__global__ void deg_scatter_kernel(const int* __restrict__ col,
                                   const float* __restrict__ ew,
                                   float* __restrict__ deg, int E) {
    int e = blockIdx.x * blockDim.x + threadIdx.x;
    if (e < E) atomicAdd(&deg[col[e]], ew[e]);
}

// 3) deg -> dinv = rsqrt(deg) (deg >= 1 always, self-loop included)
__global__ void dinv_kernel(float* __restrict__ deg, int n) {
    int i = blockIdx.x * blockDim.x + threadIdx.x;
    if (i < n) {
        float d = deg[i];
        deg[i] = (d > 0.0f) ? rsqrtf(d) : 0.0f;
    }
}

// 4) norm[e] = dinv[row]*w*dinv[col]
__global__ void norm_kernel(const int* __restrict__ row, const int* __restrict__ col,
                            const float* __restrict__ ew, const float* __restrict__ dinv,
                            float* __restrict__ nrm, int E) {
    int e = blockIdx.x * blockDim.x + threadIdx.x;
    if (e < E) nrm[e] = dinv[row[e]] * ew[e] * dinv[col[e]];
}

// ---------------------------------------------------------------------------
// GEMM: XW[N,64] = X[N,64] @ W[64,64] via V_WMMA_F32_16X16X4_F32.
// Block = 256 threads = 8 waves. Wave w: row-tile = w/4, col-tile = w%4.
// Each wave produces one 16x16 fp32 tile, K swept in steps of 4.
__global__ void __launch_bounds__(256)
gemm_wmma_kernel(const float* __restrict__ X, const float* __restrict__ W,
                 float* __restrict__ XW, int n_rows) {
    const int wave = threadIdx.x >> 5;
    const int lane = threadIdx.x & 31;
    const int row_base = blockIdx.x * 32 + (wave >> 2) * 16;
    const int col_base = (wave & 3) * 16;

    const int m     = lane & 15;          // M index (A) / N index (B) for this lane
    const int khalf = (lane >> 4) << 1;   // 0 for lanes 0-15, 2 for lanes 16-31

    int arow = row_base + m;
    if (arow >= n_rows) arow = n_rows - 1;        // safety clamp (N%32==0 normally)
    const float* __restrict__ xrow = X + (size_t)arow * DIM;

    v8f acc = {0.f, 0.f, 0.f, 0.f, 0.f, 0.f, 0.f, 0.f};

#pragma unroll
    for (int k0 = 0; k0 < DIM; k0 += 4) {
        // A 16x4: VGPR0 = K=k0+khalf, VGPR1 = K=k0+khalf+1
        v2f a;
        a.x = xrow[k0 + khalf];
        a.y = xrow[k0 + khalf + 1];
        // B 4x16: row K striped across lanes (N = lane&15), same K split as A
        v2f b;
        b.x = W[(size_t)(k0 + khalf) * DIM + col_base + m];
        b.y = W[(size_t)(k0 + khalf + 1) * DIM + col_base + m];
        // (neg_a, A, neg_b, B, c_mod, C, reuse_a, reuse_b)
        acc = __builtin_amdgcn_wmma_f32_16x16x4_f32(false, a, false, b,
                                                    (short)0, acc, false, false);
    }

    // C/D layout: VGPR v, lanes 0-15 -> M=v, lanes 16-31 -> M=v+8; N = col_base + (lane&15)
    const int m_off   = (lane >> 4) * 8;
    const int out_col = col_base + (lane & 15);
#pragma unroll
    for (int v = 0; v < 8; ++v) {
        const int r = row_base + m_off + v;
        if (r < n_rows) XW[(size_t)r * DIM + out_col] = acc[v];
    }
}

// ---------------------------------------------------------------------------
// out[n,d] = b[d] + xw[n,d] * dinv[n]^2   (bias + self-loop contribution)
__global__ void init_out_kernel(const float* __restrict__ xw, const float* __restrict__ b,
                                const float* __restrict__ dinv, float* __restrict__ out,
                                int n) {
    int idx = blockIdx.x * blockDim.x + threadIdx.x;
    if (idx < n * DIM) {
        int node = idx >> 6;
        int d    = idx & 63;
        float di = dinv[node];
        out[idx] = b[d] + xw[idx] * di * di;
    }
}

// out[col[e], d] += xw[row[e], d] * norm[e]   — 64 lanes per edge, 256B coalesced
__global__ void edge_scatter_kernel(const int* __restrict__ row, const int* __restrict__ col,
                                    const float* __restrict__ nrm, const float* __restrict__ xw,
                                    float* __restrict__ out, int E) {
    int tid = blockIdx.x * blockDim.x + threadIdx.x;
    int e = tid >> 6;
    if (e >= E) return;
    int d = tid & 63;
    int r = row[e];
    int c = col[e];
    float v = xw[(size_t)r * DIM + d] * nrm[e];
    atomicAdd(&out[(size_t)c * DIM + d], v);
}

// dst = 0.9*out + 0.1*input   (PRESERVE = 0.1)
__global__ void mix_kernel(const float* __restrict__ out, const float* __restrict__ input,
                           float* __restrict__ dst, int total) {
    int i = blockIdx.x * blockDim.x + threadIdx.x;
    if (i < total) dst[i] = 0.9f * out[i] + 0.1f * input[i];
}

// ---------------------------------------------------------------------------
extern "C" void kernel_launch(void* const* d_in, const int* in_sizes, int n_in,
                              void* d_out, int out_size, void* d_ws, size_t ws_size,
                              hipStream_t stream) {
    const float* x  = (const float*)d_in[0];   // [N,64]
    const int*   ei = (const int*)  d_in[1];   // [2,E]
    const float* ew = (const float*)d_in[2];   // [E]
    const float* W1 = (const float*)d_in[3];   // [64,64]
    const float* b1 = (const float*)d_in[4];   // [64]
    const float* W2 = (const float*)d_in[5];   // [64,64]
    const float* b2 = (const float*)d_in[6];   // [64]
    float* out = (float*)d_out;

    const int N = in_sizes[0] / DIM;
    const int E = in_sizes[2];
    const int* row = ei;
    const int* col = ei + E;

    // Workspace carve-out (256B aligned): dinv | norm | xw | obuf | tbuf
    char* ws = (char*)d_ws;
    size_t off = 0;
    auto carve = [&](size_t bytes) -> float* {
        float* p = (float*)(ws + off);
        off += (bytes + 255) & ~(size_t)255;
        return p;
    };
    float* dinv = carve((size_t)N * sizeof(float));
    float* nrm  = carve((size_t)E * sizeof(float));
    float* xw   = carve((size_t)N * DIM * sizeof(float));
    float* obuf = carve((size_t)N * DIM * sizeof(float));
    float* tbuf = carve((size_t)N * DIM * sizeof(float));

    const int TPB = 256;
    const int total = N * DIM;

    // --- edge-normalization precompute (shared by both layers) ---
    deg_init_kernel<<<(N + TPB - 1) / TPB, TPB, 0, stream>>>(dinv, N);
    deg_scatter_kernel<<<(E + TPB - 1) / TPB, TPB, 0, stream>>>(col, ew, dinv, E);
    dinv_kernel<<<(N + TPB - 1) / TPB, TPB, 0, stream>>>(dinv, N);
    norm_kernel<<<(E + TPB - 1) / TPB, TPB, 0, stream>>>(row, col, ew, dinv, nrm, E);

    const int gemm_blocks = (N + 31) / 32;
    const int scat_blocks = (int)(((size_t)E * DIM + TPB - 1) / TPB);
    const int elem_blocks = (total + TPB - 1) / TPB;

    // --- layer 1: input = x, output mix -> tbuf ---
    gemm_wmma_kernel<<<gemm_blocks, TPB, 0, stream>>>(x, W1, xw, N);
    init_out_kernel<<<elem_blocks, TPB, 0, stream>>>(xw, b1, dinv, obuf, N);
    edge_scatter_kernel<<<scat_blocks, TPB, 0, stream>>>(row, col, nrm, xw, obuf, E);
    mix_kernel<<<elem_blocks, TPB, 0, stream>>>(obuf, x, tbuf, total);

    // --- layer 2: input = tbuf, output mix -> d_out ---
    gemm_wmma_kernel<<<gemm_blocks, TPB, 0, stream>>>(tbuf, W2, xw, N);
    init_out_kernel<<<elem_blocks, TPB, 0, stream>>>(xw, b2, dinv, obuf, N);
    edge_scatter_kernel<<<scat_blocks, TPB, 0, stream>>>(row, col, nrm, xw, obuf, E);
    mix_kernel<<<elem_blocks, TPB, 0, stream>>>(obuf, tbuf, out, total);
}